// ramsey_MPNN_41463614276026
// MI455X (gfx1250) — compile-verified
//
#include <hip/hip_runtime.h>
#include <hip/hip_bf16.h>

typedef float v2f __attribute__((ext_vector_type(2)));
typedef float v8f __attribute__((ext_vector_type(8)));

#define NUM_FEAT    64
#define HIDDEN      128
#define N_CLIQUES   500000
#define TILES_PER_SET (N_CLIQUES / 16)      // 31250
#define TOTAL_TILES   (2 * TILES_PER_SET)   // 62500

__global__ __launch_bounds__(256) void ramsey_edge_pred_kernel(
    const float* __restrict__ node_features,
    const int*   __restrict__ cliques_r,
    const int*   __restrict__ cliques_s,
    const float* __restrict__ W5,
    const float* __restrict__ b5,
    const float* __restrict__ W6,
    const float* __restrict__ b6,
    float*       __restrict__ out)
{
    // W5 staged in WMMA B-operand order: [nt][kk][lane] -> float2 (32 KB of 320 KB LDS)
    __shared__ v2f sW5[8 * 16 * 32];

    const int tid = threadIdx.x;
    for (int t = tid; t < 8 * 16 * 32; t += 256) {
        int lane = t & 31;
        int kk   = (t >> 5) & 15;
        int nt   = t >> 9;
        int k0 = 4 * kk + ((lane >> 4) << 1);   // K rows this lane supplies
        int n  = 16 * nt + (lane & 15);         // N column this lane supplies
        v2f w;
        w.x = W5[k0 * HIDDEN + n];
        w.y = W5[(k0 + 1) * HIDDEN + n];
        sW5[t] = w;
    }
    __syncthreads();

    const int lane = tid & 31;
    const int warp = tid >> 5;
    const int half = lane >> 4;   // 0: rows M 0..7 / K pair {0,1};  1: rows M 8..15 / K pair {2,3}
    const int sub  = lane & 15;

    const int tile = blockIdx.x * 8 + warp;
    if (tile >= TOTAL_TILES) return;            // uniform per-wave exit (EXEC stays all-ones)

    const int set  = (tile >= TILES_PER_SET) ? 1 : 0;
    const int tin  = tile - set * TILES_PER_SET;
    const int* __restrict__ cliques = set ? cliques_s : cliques_r;
    float* __restrict__ outp = out + (size_t)set * (2 * N_CLIQUES);

    const int e0 = tin * 16;

    // Edge index gather: lanes m and m+16 both service edge m's row (cols 0,1 of the clique)
    const int eb = (e0 + sub) * 4;
    const int i0 = cliques[eb + 0];
    const int i1 = cliques[eb + 1];
    const float* __restrict__ rowi = node_features + (size_t)i0 * NUM_FEAT;
    const float* __restrict__ rowj = node_features + (size_t)i1 * NUM_FEAT;

    // Gather + elementwise multiply straight into A-operand registers (L2-resident gathers)
    v2f areg[16];
#pragma unroll
    for (int kk = 0; kk < 16; ++kk) {
        int k0 = 4 * kk + (half << 1);
        v2f xi = *(const v2f*)(rowi + k0);
        v2f xj = *(const v2f*)(rowj + k0);
        v2f a;
        a.x = xi.x * xj.x;
        a.y = xi.y * xj.y;
        areg[kk] = a;
    }

    float part[8][2];
#pragma unroll
    for (int r = 0; r < 8; ++r) { part[r][0] = 0.f; part[r][1] = 0.f; }

    for (int nt = 0; nt < 8; ++nt) {
        // accumulator pre-loaded with b5 broadcast (bias is N-only, identical across the 8 M-rows)
        float bias = b5[16 * nt + sub];
        v8f acc = { bias, bias, bias, bias, bias, bias, bias, bias };

        const v2f* __restrict__ bptr = &sW5[(nt * 16) * 32 + lane];
#pragma unroll
        for (int kk = 0; kk < 16; ++kk) {
            v2f bop = bptr[kk * 32];            // conflict-free ds_load_b64
            acc = __builtin_amdgcn_wmma_f32_16x16x4_f32(
                false, areg[kk], false, bop, (short)0, acc, false, false);
        }

        // ReLU and fold this N-tile into the 128-wide logits dot (W6: 128x2)
        v2f w6 = *(const v2f*)(W6 + (16 * nt + sub) * 2);
#pragma unroll
        for (int r = 0; r < 8; ++r) {
            float h = acc[r] > 0.f ? acc[r] : 0.f;
            part[r][0] = fmaf(h, w6.x, part[r][0]);
            part[r][1] = fmaf(h, w6.y, part[r][1]);
        }
    }

    // Butterfly-reduce the N dimension across each 16-lane half (masks 1,2,4,8 stay in-half)
#pragma unroll
    for (int mask = 1; mask <= 8; mask <<= 1) {
#pragma unroll
        for (int r = 0; r < 8; ++r) {
            part[r][0] += __shfl_xor(part[r][0], mask, 32);
            part[r][1] += __shfl_xor(part[r][1], mask, 32);
        }
    }

    // 16 lanes publish 16 edges: lane (sub<8) of each half owns edge m = sub + 8*half
    if (sub < 8) {
        const int m = sub + 8 * half;
        float l0 = part[sub][0] + b6[0];
        float l1 = part[sub][1] + b6[1];
        float mx = fmaxf(l0, l1);
        float z0 = __expf(l0 - mx);
        float z1 = __expf(l1 - mx);
        float inv = 1.f / (z0 + z1);
        v2f p;
        p.x = z0 * inv;
        p.y = z1 * inv;
        *(v2f*)(outp + (size_t)(e0 + m) * 2) = p;
    }
}

extern "C" void kernel_launch(void* const* d_in, const int* in_sizes, int n_in,
                              void* d_out, int out_size, void* d_ws, size_t ws_size,
                              hipStream_t stream) {
    (void)in_sizes; (void)n_in; (void)d_ws; (void)ws_size; (void)out_size;
    // setup_inputs order: x(unused), cliques_r, cliques_s, node_features, W5, b5, W6, b6
    const int*   cliques_r     = (const int*)  d_in[1];
    const int*   cliques_s     = (const int*)  d_in[2];
    const float* node_features = (const float*)d_in[3];
    const float* W5            = (const float*)d_in[4];
    const float* b5            = (const float*)d_in[5];
    const float* W6            = (const float*)d_in[6];
    const float* b6            = (const float*)d_in[7];
    float* out = (float*)d_out;

    const int blocks = (TOTAL_TILES + 7) / 8;   // 8 waves (tiles) per 256-thread block
    ramsey_edge_pred_kernel<<<blocks, 256, 0, stream>>>(
        node_features, cliques_r, cliques_s, W5, b5, W6, b6, out);
}